// Graph_Session_74431783239701
// MI455X (gfx1250) — compile-verified
//
#include <hip/hip_runtime.h>
#include <hip/hip_bf16.h>

#define EPS_BN 1e-5f
#define LREL   0.2f
#define Bs   512
#define Ls   50
#define Ds   128
#define Hs   4
#define Vs   100000
#define BTs  4096

typedef __attribute__((ext_vector_type(2))) float v2f;
typedef __attribute__((ext_vector_type(8))) float v8f;

// ---------------------------------------------------------------------------
// f32 WMMA GEMM: C[M,N] = act(A[M,K] * B + bias)
//   BT_=0: B is [K,N] row-major.  BT_=1: B is [N,K] row-major (A @ B^T).
//   One wave -> one 16x16 C tile, 4 waves per block over consecutive N tiles.
//   V_WMMA_F32_16X16X4_F32; K must be a multiple of 8, M a multiple of 16.
//   A 16x4 frag: lane m = lane&15, VGPR{0,1} = K = 2*(lane>>4) + {0,1}.
//   B 4x16 frag: lane n = lane&15, same K mapping (mirrors C half-lane split).
//   C/D: VGPR r -> row m0 + r + 8*(lane>>4), col n0 + (lane&15).
// ACT_: 0 = none, 1 = elu.  Template params keep the inner loop branch-free.
// ---------------------------------------------------------------------------
template<int BT_, int ACT_>
__global__ void gemm16_kernel(const float* __restrict__ A, const float* __restrict__ B,
                              const float* __restrict__ bias, float* __restrict__ C,
                              int M, int N, int K, int lda, int ldb, int ldc)
{
    int lane = threadIdx.x & 31;
    int wid  = threadIdx.x >> 5;
    int half = lane >> 4;
    int lq   = lane & 15;
    int m0   = blockIdx.y * 16;
    int n0   = (blockIdx.x * 4 + wid) * 16;
    if (n0 >= N) return;                 // wave-uniform: EXEC stays all-ones for WMMA

    v8f acc = {};
    const float* Arow = A + (size_t)(m0 + lq) * lda;
    int nB = n0 + lq;                    // for BT_=1 this row always exists (u2e has V rows)
    const float* Brow = BT_ ? (B + (size_t)nB * ldb) : nullptr;
    if (BT_) __builtin_prefetch(Brow, 0, 1);   // global_prefetch_b8: stream u2e row ahead

    for (int k = 0; k < K; k += 8) {
        int kk = k + half * 2;
        v2f a0 = *(const v2f*)(Arow + kk);
        v2f a1 = *(const v2f*)(Arow + kk + 4);
        v2f b0, b1;
        if (BT_) {
            b0 = *(const v2f*)(Brow + kk);
            b1 = *(const v2f*)(Brow + kk + 4);
        } else {
            b0.x = B[(size_t)kk * ldb + nB];
            b0.y = B[(size_t)(kk + 1) * ldb + nB];
            b1.x = B[(size_t)(kk + 4) * ldb + nB];
            b1.y = B[(size_t)(kk + 5) * ldb + nB];
        }
        acc = __builtin_amdgcn_wmma_f32_16x16x4_f32(false, a0, false, b0,
                                                    (short)0, acc, false, false);
        acc = __builtin_amdgcn_wmma_f32_16x16x4_f32(false, a1, false, b1,
                                                    (short)0, acc, false, false);
    }
    int col = n0 + lq;
    if (col < N) {
        float bv = bias ? bias[col] : 0.f;
#pragma unroll
        for (int r = 0; r < 8; ++r) {
            int row = m0 + r + half * 8;
            float v = acc[r] + bv;
            if (ACT_ == 1) v = v > 0.f ? v : (__expf(v) - 1.f);
            C[(size_t)row * ldc + col] = v;
        }
    }
}

// ---------------- gather rows of a [*,128] f32 table ----------------
__global__ void gather_kernel(const float* __restrict__ table, const int* __restrict__ idx,
                              float* __restrict__ out, int n, int ldo)
{
    int t = blockIdx.x * blockDim.x + threadIdx.x;
    if (t >= n * 32) return;
    int i = t >> 5, q = t & 31;
    const float4* src = (const float4*)(table + (size_t)idx[i] * Ds);
    float4* dst = (float4*)(out + (size_t)i * ldo);
    dst[q] = src[q];
}

// ---------------- batchnorm: column mean/var over M rows ----------------
__global__ void colstats_kernel(const float* __restrict__ X, int M, int C,
                                float* __restrict__ mv)
{
    int c = blockIdx.x, t = threadIdx.x;
    __shared__ float ss[256], sq[256];
    float s = 0.f, q = 0.f;
    for (int r = t; r < M; r += 256) {
        float v = X[(size_t)r * C + c];
        s += v; q += v * v;
    }
    ss[t] = s; sq[t] = q; __syncthreads();
    for (int off = 128; off > 0; off >>= 1) {
        if (t < off) { ss[t] += ss[t + off]; sq[t] += sq[t + off]; }
        __syncthreads();
    }
    if (t == 0) {
        float m = ss[0] / (float)M;
        mv[c] = m;
        mv[C + c] = sq[0] / (float)M - m * m;
    }
}

__global__ void bn_relu_kernel(float* __restrict__ X, int n, int C,
                               const float* __restrict__ mv,
                               const float* __restrict__ g, const float* __restrict__ b)
{
    int i = blockIdx.x * blockDim.x + threadIdx.x;
    if (i >= n) return;
    int c = i % C;
    float y = (X[i] - mv[c]) * rsqrtf(mv[C + c] + EPS_BN) * g[c] + b[c];
    X[i] = fmaxf(y, 0.f);
}

// ---------------- GAT: decomposed attention, masked softmax, aggregate -----
__global__ void gat_kernel(const float* __restrict__ hin,
                           const float* __restrict__ a1, const float* __restrict__ a2,
                           const int* __restrict__ mask,
                           float* __restrict__ out, int ldo, int H, int do_elu)
{
    __shared__ float hs[Ls * Ds];
    __shared__ float s1s[Hs * Ls], s2s[Hs * Ls], mk[Ls];
    int b = blockIdx.x;
    const float* hb = hin + (size_t)b * Ls * Ds;
    for (int t = threadIdx.x; t < Ls * Ds; t += blockDim.x) hs[t] = hb[t];
    for (int t = threadIdx.x; t < Ls; t += blockDim.x) mk[t] = (float)mask[b * Ls + t];
    __syncthreads();
    for (int t = threadIdx.x; t < H * Ls; t += blockDim.x) {
        int hd = t / Ls, i = t % Ls;
        const float* a1p = a1 + hd * Ds;
        const float* a2p = a2 + hd * Ds;
        float d1 = 0.f, d2 = 0.f;
        for (int dd = 0; dd < Ds; ++dd) {
            float hv = hs[i * Ds + dd];
            d1 += hv * a1p[dd]; d2 += hv * a2p[dd];
        }
        s1s[t] = d1; s2s[t] = d2;
    }
    __syncthreads();
    int wid = threadIdx.x >> 5, lane = threadIdx.x & 31, nw = blockDim.x >> 5;
    for (int t = wid; t < H * Ls; t += nw) {
        int hd = t / Ls, i = t % Ls;
        float si = s1s[hd * Ls + i];
        float mx = -1e30f;
        for (int j = 0; j < Ls; ++j) {
            float e = si + s2s[hd * Ls + j];
            e = e > 0.f ? e : LREL * e;
            e = (mk[j] != 0.f) ? e : -1e9f;
            mx = fmaxf(mx, e);
        }
        float sum = 0.f;
        float4 acc = make_float4(0.f, 0.f, 0.f, 0.f);
        for (int j = 0; j < Ls; ++j) {
            float e = si + s2s[hd * Ls + j];
            e = e > 0.f ? e : LREL * e;
            e = (mk[j] != 0.f) ? e : -1e9f;
            float p = __expf(e - mx);
            sum += p;
            float4 hj = *(const float4*)&hs[j * Ds + lane * 4];
            acc.x += p * hj.x; acc.y += p * hj.y; acc.z += p * hj.z; acc.w += p * hj.w;
        }
        float inv = 1.f / sum;
        float4 o; o.x = acc.x * inv; o.y = acc.y * inv; o.z = acc.z * inv; o.w = acc.w * inv;
        if (do_elu) {
            o.x = o.x > 0.f ? o.x : __expf(o.x) - 1.f;
            o.y = o.y > 0.f ? o.y : __expf(o.y) - 1.f;
            o.z = o.z > 0.f ? o.z : __expf(o.z) - 1.f;
            o.w = o.w > 0.f ? o.w : __expf(o.w) - 1.f;
        }
        *(float4*)(out + ((size_t)b * Ls + i) * ldo + hd * Ds + lane * 4) = o;
    }
}

// ---------------- compute_scores glue kernels ----------------
__global__ void seqht_kernel(const float* __restrict__ hidden, const int* __restrict__ mask,
                             float* __restrict__ acat)
{
    int b = blockIdx.x, d = threadIdx.x;
    __shared__ int sl;
    if (d == 0) { int s = 0; for (int l = 0; l < Ls; ++l) s += mask[b * Ls + l]; sl = s; }
    __syncthreads();
    acat[b * 256 + 128 + d] = hidden[((size_t)b * Ls + (sl - 1)) * Ds + d];
}

__global__ void alpha_a_kernel(const float* __restrict__ q1, const float* __restrict__ q2,
                               const float* __restrict__ hidden, const int* __restrict__ mask,
                               const float* __restrict__ W3, float* __restrict__ acat)
{
    int b = blockIdx.x, d = threadIdx.x;
    __shared__ float red[128];
    float q1d = q1[b * Ds + d];
    float w3d = W3[d];
    float acc = 0.f;
    for (int l = 0; l < Ls; ++l) {
        float s = q1d + q2[((size_t)b * Ls + l) * Ds + d];
        s = 1.f / (1.f + __expf(-s));
        red[d] = s * w3d;
        __syncthreads();
        for (int off = 64; off > 0; off >>= 1) {
            if (d < off) red[d] += red[d + off];
            __syncthreads();
        }
        float alpha = red[0];
        __syncthreads();
        float mf = (float)mask[b * Ls + l];
        acc += alpha * hidden[((size_t)b * Ls + l) * Ds + d] * mf;
    }
    acat[b * 256 + d] = acc;
}

__global__ void pmax_kernel(const float* __restrict__ h0, const int* __restrict__ mask,
                            float* __restrict__ pcat)
{
    int b = blockIdx.x, d = threadIdx.x;
    float m = -1e30f;
    for (int l = 0; l < Ls; ++l) {
        float v = mask[b * Ls + l] ? h0[((size_t)b * Ls + l) * Ds + d] : 0.f;
        m = fmaxf(m, v);
    }
    pcat[b * 256 + 128 + d] = m;
}

__global__ void av_kernel(const float* __restrict__ pcat, const float* __restrict__ att_t,
                          float* __restrict__ av)
{
    int b = blockIdx.x, d = threadIdx.x;
    __shared__ float r0[128], r1[128];
    float pa = pcat[b * 256 + d], pm = pcat[b * 256 + 128 + d];
    r0[d] = pa * att_t[d * 2 + 0] + pm * att_t[(128 + d) * 2 + 0];
    r1[d] = pa * att_t[d * 2 + 1] + pm * att_t[(128 + d) * 2 + 1];
    __syncthreads();
    for (int off = 64; off > 0; off >>= 1) {
        if (d < off) { r0[d] += r0[d + off]; r1[d] += r1[d + off]; }
        __syncthreads();
    }
    float s0 = r0[0], s1 = r1[0];
    float mx = fmaxf(s0, s1);
    float e0 = __expf(s0 - mx), e1 = __expf(s1 - mx);
    float g0 = e0 / (e0 + e1), g1 = e1 / (e0 + e1);
    av[b * Ds + d] = pa * g0 + pm * g1;
}

__global__ void tower_final_kernel(const float* __restrict__ x, const float* __restrict__ x0,
                                   const float* __restrict__ gate_w,
                                   const float* __restrict__ Wuv3, const float* __restrict__ buv3,
                                   float* __restrict__ out)
{
    int r = blockIdx.x * blockDim.x + threadIdx.x;
    if (r >= BTs) return;
    float s0 = 0.f, s1 = 0.f;
    for (int c = 0; c < 16; ++c) { float v = x[r * 16 + c];  s0 += v * gate_w[c * 2];        s1 += v * gate_w[c * 2 + 1]; }
    for (int c = 0; c < 16; ++c) { float v = x0[r * 16 + c]; s0 += v * gate_w[(16 + c) * 2]; s1 += v * gate_w[(16 + c) * 2 + 1]; }
    float mx = fmaxf(s0, s1);
    float e0 = __expf(s0 - mx), e1 = __expf(s1 - mx);
    float g0 = e0 / (e0 + e1), g1 = e1 / (e0 + e1);
    float sc = buv3[0];
    for (int c = 0; c < 16; ++c)
        sc += (g0 * x[r * 16 + c] + g1 * x0[r * 16 + c]) * Wuv3[c];
    out[r] = sc;
}

// ---------------------------------------------------------------------------
// input index map (setup_inputs dict order, params flattened first)
enum {
    P_U2E = 0, P_USER_EMB, P_V2E, P_W1, P_B1, P_W2, P_B2, P_W3, P_WT, P_BTv,
    P_ATT_T, P_ATT_A1, P_ATT_A2, P_OUT_A1, P_OUT_A2, P_W,
    P_WUR1, P_BUR1, P_WUR2, P_BUR2, P_WVR1, P_BVR1, P_WVR2, P_BVR2,
    P_WUV1, P_BUV1, P_WUV2, P_BUV2, P_WUV3, P_BUV3,
    P_WUV10, P_BUV10, P_WUV20, P_BUV20, P_GATE_W,
    P_G1, P_BE1, P_G2, P_BE2, P_G3, P_BE3, P_G4, P_BE4,
    P_G30, P_BE30, P_G40, P_BE40,
    I_NODES_U, I_NODES_V, I_INPUTS, I_MASK
};

extern "C" void kernel_launch(void* const* d_in, const int* in_sizes, int n_in,
                              void* d_out, int out_size, void* d_ws, size_t ws_size,
                              hipStream_t stream)
{
    (void)in_sizes; (void)n_in; (void)out_size; (void)ws_size;
    auto F = [&](int i) { return (const float*)d_in[i]; };
    const int* nodes_u = (const int*)d_in[I_NODES_U];
    const int* nodes_v = (const int*)d_in[I_NODES_V];
    const int* inputs  = (const int*)d_in[I_INPUTS];
    const int* mask    = (const int*)d_in[I_MASK];
    float* out = (float*)d_out;
    float* ws  = (float*)d_ws;

    // ---- workspace layout (floats); peak ~64 MiB with aliasing ----
    const size_t OF_H0  = 0;                         // [25600,128] h0 ; later mul_one
    const size_t OF_BIG = (size_t)Bs * Ls * Ds;      // 3,276,800 : tower / mul_seq / hidden+q2
    const size_t BIGSZ  = (size_t)Bs * Ls * Hs * Ds; // 13,107,200
    float* h0      = ws + OF_H0;
    float* mul_one = ws + OF_H0;                     // alias (h0 dead after GAT1+pmax)
    float* mul_seq = ws + OF_BIG;
    float* hidden  = ws + OF_BIG;                    // alias (mul_seq dead after mul_one)
    float* q2      = ws + OF_BIG + (size_t)Bs * Ls * Ds;
    // tower buffers live inside OF_BIG region (tower runs before mul_seq)
    float* euv  = ws + OF_BIG;                       // [4096,256]
    float* t1   = euv  + (size_t)BTs * 256;          // [4096,128]
    float* xuxv = t1   + (size_t)BTs * 128;          // [4096,256]
    float* y1   = xuxv + (size_t)BTs * 256;          // [4096,128]
    float* xbuf = y1   + (size_t)BTs * 128;          // [4096,16]
    float* y10  = xbuf + (size_t)BTs * 16;           // [4096,128]
    float* x0b  = y10  + (size_t)BTs * 128;          // [4096,16]
    // small region
    float* acat  = ws + OF_BIG + BIGSZ;              // [512,256] = [a | ht]
    float* pcat  = acat + (size_t)Bs * 256;          // [512,256] = [p_a | p_max]
    float* av    = pcat + (size_t)Bs * 256;          // [512,128]
    float* q1    = av   + (size_t)Bs * 128;          // [512,128]
    float* stats = q1   + (size_t)Bs * 128;          // [512] mean||var

    auto gemm = [&](const float* A, const float* B, const float* bias, float* C,
                    int M, int N, int K, int lda, int ldb, int ldc, int bT, int act) {
        int ntiles = (N + 15) / 16;
        dim3 g((ntiles + 3) / 4, (M + 15) / 16);
        if (bT)
            gemm16_kernel<1, 0><<<g, 128, 0, stream>>>(A, B, bias, C, M, N, K, lda, ldb, ldc);
        else if (act)
            gemm16_kernel<0, 1><<<g, 128, 0, stream>>>(A, B, bias, C, M, N, K, lda, ldb, ldc);
        else
            gemm16_kernel<0, 0><<<g, 128, 0, stream>>>(A, B, bias, C, M, N, K, lda, ldb, ldc);
    };
    auto bnrelu = [&](float* X, int M, int C, const float* g, const float* b) {
        colstats_kernel<<<C, 256, 0, stream>>>(X, M, C, stats);
        int n = M * C;
        bn_relu_kernel<<<(n + 255) / 256, 256, 0, stream>>>(X, n, C, stats, g, b);
    };

    // ---- gathers ----
    gather_kernel<<<(BTs * 32 + 255) / 256, 256, 0, stream>>>(F(P_USER_EMB), nodes_u, euv, BTs, 256);
    gather_kernel<<<(BTs * 32 + 255) / 256, 256, 0, stream>>>(F(P_V2E), nodes_v, euv + 128, BTs, 256);
    gather_kernel<<<(Bs * Ls * 32 + 255) / 256, 256, 0, stream>>>(F(P_U2E), inputs, h0, Bs * Ls, Ds);

    // ---- rating towers ----
    gemm(euv,        F(P_WUR1), F(P_BUR1), t1,         BTs, 128, 128, 256, 128, 128, 0, 0);
    bnrelu(t1, BTs, 128, F(P_G1), F(P_BE1));
    gemm(t1,         F(P_WUR2), F(P_BUR2), xuxv,       BTs, 128, 128, 128, 128, 256, 0, 0);
    gemm(euv + 128,  F(P_WVR1), F(P_BVR1), t1,         BTs, 128, 128, 256, 128, 128, 0, 0);
    bnrelu(t1, BTs, 128, F(P_G2), F(P_BE2));
    gemm(t1,         F(P_WVR2), F(P_BVR2), xuxv + 128, BTs, 128, 128, 128, 128, 256, 0, 0);
    gemm(xuxv,       F(P_WUV1), F(P_BUV1), y1,         BTs, 128, 256, 256, 128, 128, 0, 0);
    bnrelu(y1, BTs, 128, F(P_G3), F(P_BE3));
    gemm(y1,         F(P_WUV2), F(P_BUV2), xbuf,       BTs, 16,  128, 128, 16,  16,  0, 0);
    bnrelu(xbuf, BTs, 16, F(P_G4), F(P_BE4));
    gemm(euv,        F(P_WUV10), F(P_BUV10), y10,      BTs, 128, 256, 256, 128, 128, 0, 0);
    bnrelu(y10, BTs, 128, F(P_G30), F(P_BE30));
    gemm(y10,        F(P_WUV20), F(P_BUV20), x0b,      BTs, 16,  128, 128, 16,  16,  0, 0);
    bnrelu(x0b, BTs, 16, F(P_G40), F(P_BE40));
    tower_final_kernel<<<(BTs + 255) / 256, 256, 0, stream>>>(
        xbuf, x0b, F(P_GATE_W), F(P_WUV3), F(P_BUV3), out);

    // ---- p_max (only depends on h0; do before h0 slot is reused) ----
    pmax_kernel<<<Bs, 128, 0, stream>>>(h0, mask, pcat);

    // ---- session GAT ----
    gat_kernel<<<Bs, 256, 0, stream>>>(h0, F(P_ATT_A1), F(P_ATT_A2), mask,
                                       mul_seq, Hs * Ds, Hs, 1);
    gemm(mul_seq, F(P_W), nullptr, mul_one, Bs * Ls, 128, Hs * Ds, Hs * Ds, 128, 128, 0, 1); // elu
    gat_kernel<<<Bs, 256, 0, stream>>>(mul_one, F(P_OUT_A1), F(P_OUT_A2), mask,
                                       hidden, Ds, 1, 0);

    // ---- compute_scores ----
    seqht_kernel<<<Bs, 128, 0, stream>>>(hidden, mask, acat);               // ht -> acat[:,128:]
    gemm(acat + 128, F(P_W1), F(P_B1), q1, Bs, 128, 128, 256, 128, 128, 0, 0);
    gemm(hidden,     F(P_W2), F(P_B2), q2, Bs * Ls, 128, 128, 128, 128, 128, 0, 0);
    alpha_a_kernel<<<Bs, 128, 0, stream>>>(q1, q2, hidden, mask, F(P_W3), acat); // a -> acat[:,:128]
    gemm(acat, F(P_WT), F(P_BTv), pcat, Bs, 128, 256, 256, 128, 256, 0, 0);      // p_a -> pcat[:,:128]
    av_kernel<<<Bs, 128, 0, stream>>>(pcat, F(P_ATT_T), av);
    // scores = av @ u2e[:-1]^T  -> out[BT:], write-bandwidth bound (~205 MB)
    gemm(av, F(P_U2E), nullptr, out + BTs, Bs, Vs - 1, 128, 128, 128, Vs - 1, 1, 0);
}